// RoIAlign_24790551232870
// MI455X (gfx1250) — compile-verified
//
#include <hip/hip_runtime.h>

// RoI Align forward for MI455X (gfx1250, wave32).
//
// Gather-bound op (~0.5 FLOP/byte): the 82 MB input is L2-resident (192 MB L2),
// HBM floor ~ (82+50)MB / 23.3 TB/s ~ 5.7 us. No matmul structure -> no WMMA.
// Strategy:
//   * one 256-thread block (8 wave32s) per roi; lane = channel
//   * per-roi (offset, weight) tap table built once in LDS, shared by all C
//   * x-taps are memory-adjacent: each bilinear sample = 2 paired-row loads
//     (global_load_b64) instead of 4 scalar loads -> 392 gathers/thread
//   * global_prefetch_b8 warms the next bin's cache lines

namespace {
constexpr int PH = 7, PW = 7, SR = 2;
constexpr int NBINS   = PH * PW;           // 49
constexpr int SAMP_X  = PW * SR;           // 14
constexpr int NSAMP   = SAMP_X * SAMP_X;   // 196 sample points per roi
constexpr int SPB     = SR * SR;           // 4 samples per bin
constexpr int Cc = 256, Hh = 200, Ww = 200;
constexpr float SCALE = 0.25f;
}

// two adjacent pixels; natural align 4 so a 4B-aligned paired load is legal
struct F2 { float lo, hi; };

__global__ __launch_bounds__(256)
void roi_align_fwd_kernel(const float* __restrict__ input,
                          const float* __restrict__ rois,
                          float* __restrict__ out)
{
    __shared__ int2   s_off[NSAMP];        // {offTop, offBot} per sample, 1568 B
    __shared__ float4 s_w  [NSAMP];        // {wTL, wTH, wBL, wBH},       3136 B
    __shared__ int planeBase;

    const int k   = blockIdx.x;
    const int tid = threadIdx.x;

    // ---------------- Phase 1: build paired-tap table ----------------------
    if (tid < NSAMP) {
        const float x1 = rois[k * 5 + 1] * SCALE;
        const float y1 = rois[k * 5 + 2] * SCALE;
        const float x2 = rois[k * 5 + 3] * SCALE;
        const float y2 = rois[k * 5 + 4] * SCALE;
        const float binW = fmaxf(x2 - x1, 1.0f) * (1.0f / PW);
        const float binH = fmaxf(y2 - y1, 1.0f) * (1.0f / PH);

        const int ix = tid % SAMP_X;       // 0..13 sample column
        const int iy = tid / SAMP_X;       // 0..13 sample row
        // sample coord: x1 + (pw + (s+0.5)/sr) * bin_w ; sr==2 -> 0.25 + 0.5*s
        const float xs = x1 + ((float)(ix >> 1) + 0.25f + 0.5f * (float)(ix & 1)) * binW;
        const float ys = y1 + ((float)(iy >> 1) + 0.25f + 0.5f * (float)(iy & 1)) * binH;

        // _interp_axis(ys, H): rows yT (weight 1-fy) and yB (weight fy)
        const bool vy = (ys >= -1.0f) && (ys <= (float)Hh);
        const float cy = fmaxf(ys, 0.0f);
        const int  yl0   = (int)cy;               // floor, cy >= 0
        const bool overY = (yl0 >= Hh - 1);
        const int  yT = overY ? (Hh - 1) : yl0;
        const int  yB = overY ? (Hh - 1) : yl0 + 1;
        const float fy = overY ? 0.0f : cy - (float)yl0;

        // _interp_axis(xs, W), remapped onto an adjacent pair at base bx:
        //   normal: elems (bx, bx+1) carry (1-fx, fx)
        //   over  : xl==xh==W-1 and fx==0 -> pair base W-2, weights (0, 1)
        const bool vx = (xs >= -1.0f) && (xs <= (float)Ww);
        const float cx = fmaxf(xs, 0.0f);
        const int  xl0   = (int)cx;
        const bool overX = (xl0 >= Ww - 1);
        const int  bx = overX ? (Ww - 2) : xl0;
        const float fx = overX ? 0.0f : cx - (float)xl0;
        const float wx0 = overX ? 0.0f : 1.0f - fx;
        const float wx1 = overX ? 1.0f : fx;

        // validity mask and 1/(sr*sr) folded into the weights
        const float m   = (vy && vx) ? 0.25f : 0.0f;
        const float wyT = m * (1.0f - fy), wyB = m * fy;

        const int bin  = (iy >> 1) * PW + (ix >> 1);
        const int samp = ((iy & 1) << 1) | (ix & 1);
        const int sm   = bin * SPB + samp;
        s_off[sm] = make_int2(yT * Ww + bx, yB * Ww + bx);
        s_w[sm]   = make_float4(wyT * wx0, wyT * wx1, wyB * wx0, wyB * wx1);
    }
    if (tid == 0) planeBase = (int)rois[k * 5] * (Cc * Hh * Ww);
    __syncthreads();

    // ---------------- Phase 2: lane = channel, stream 49 bins --------------
    const int c = tid;                     // C == blockDim.x == 256
    const float* __restrict__ plane =
        input + (ptrdiff_t)planeBase + (ptrdiff_t)c * (Hh * Ww);
    float* __restrict__ op = out + ((ptrdiff_t)k * Cc + c) * NBINS;

    #pragma unroll 1
    for (int bin = 0; bin < NBINS; ++bin) {
        if (bin + 1 < NBINS) {
            // gfx1250 global_prefetch_b8: warm next bin's cache lines
            __builtin_prefetch(plane + s_off[(bin + 1) * SPB + 0].x, 0, 1);
            __builtin_prefetch(plane + s_off[(bin + 1) * SPB + 3].y, 0, 1);
        }
        float acc = 0.0f;
        #pragma unroll
        for (int s = 0; s < SPB; ++s) {    // 8 independent paired gathers/bin
            const int2   o = s_off[bin * SPB + s];
            const float4 w = s_w  [bin * SPB + s];
            const F2 t = *(const F2*)(plane + o.x);   // top row pair
            const F2 b = *(const F2*)(plane + o.y);   // bottom row pair
            acc = fmaf(w.x, t.lo, acc);
            acc = fmaf(w.y, t.hi, acc);
            acc = fmaf(w.z, b.lo, acc);
            acc = fmaf(w.w, b.hi, acc);
        }
        op[bin] = acc;
    }
}

extern "C" void kernel_launch(void* const* d_in, const int* in_sizes, int n_in,
                              void* d_out, int out_size, void* d_ws, size_t ws_size,
                              hipStream_t stream)
{
    const float* input = (const float*)d_in[0];   // (N=2, C=256, H=200, W=200) f32
    const float* rois  = (const float*)d_in[1];   // (K, 5) f32
    float* out = (float*)d_out;                   // (K, 256, 7, 7) f32
    const int K = in_sizes[1] / 5;
    if (K <= 0) return;
    roi_align_fwd_kernel<<<dim3((unsigned)K), dim3(256), 0, stream>>>(input, rois, out);
}